// AngularPenaltySMLoss_71640054497678
// MI455X (gfx1250) — compile-verified
//
#include <hip/hip_runtime.h>
#include <math.h>

// ---- CDNA5 WMMA types ----
typedef float v2f __attribute__((ext_vector_type(2)));
typedef float v8f __attribute__((ext_vector_type(8)));

static constexpr int   D       = 256;   // feature dim
static constexpr int   ROWS    = 32;    // rows (samples) per workgroup
static constexpr int   PAD     = 2;     // LDS row padding (floats) -> conflict-free b64
static constexpr int   LDA     = D + PAD;
static constexpr float S_SCALE = 30.0f;
static constexpr float MARGIN  = 0.3f;

// Fused: row-normalize x, GEMM xn @ W^T via V_WMMA_F32_16X16X4_F32,
// online sum of exp(S*logit) per row + target logit capture. Never
// materializes the [N,C] logits (saves ~655MB of HBM traffic round-trip).
__global__ __launch_bounds__(256)
void arcface_fused(const float* __restrict__ x,
                   const float* __restrict__ Wm,
                   const int*   __restrict__ target,
                   float*       __restrict__ rowOut,
                   int C)
{
    __shared__ float aLds[ROWS * LDA];     // normalized x tile, padded
    __shared__ float rowSumP[4][ROWS];     // per-phase partial exp sums (deterministic, no atomics)
    __shared__ float tgtVal[ROWS];         // captured target-class logit
    __shared__ int   tgtIdx[ROWS];

    const int t        = threadIdx.x;
    const int rowBaseG = blockIdx.x * ROWS;

    // ---- stage x tile into LDS (float4 coalesced) ----
    {
        const float4* src = (const float4*)(x + (size_t)rowBaseG * D);
        const int nVec4 = ROWS * D / 4;                  // 2048
        for (int i = t; i < nVec4; i += 256) {
            float4 v = src[i];
            int flat = i * 4;
            int r = flat >> 8;                           // /D
            int c = flat & (D - 1);
            float* dst = &aLds[r * LDA + c];
            dst[0] = v.x; dst[1] = v.y; dst[2] = v.z; dst[3] = v.w;
        }
        if (t < ROWS) tgtIdx[t] = target[rowBaseG + t];
    }
    __syncthreads();

    // ---- L2-normalize each row in LDS (W is intentionally NOT normalized) ----
    if (t < ROWS) {
        float* rp = &aLds[t * LDA];
        float ss = 0.0f;
        for (int k = 0; k < D; ++k) { float v = rp[k]; ss += v * v; }
        float rn = 1.0f / sqrtf(ss);
        for (int k = 0; k < D; ++k) rp[k] *= rn;
    }
    __syncthreads();

    // ---- wave decomposition ----
    const int lane = t & 31;
    const int wave = t >> 5;
    const int h    = lane >> 4;        // half-wave (K/N split per ISA layout)
    const int ln   = lane & 15;        // M for A-frag, N for B/D-frag
    const int rt   = wave >> 2;        // row tile 0..1 (16 rows each)
    const int q    = wave & 3;         // column-chunk phase 0..3
    const int rowBaseL = rt * 16;
    const int chunks   = C >> 4;       // 625

    // A-frag source: lane ln = M, VGPR pair covers K = {4s+2h, 4s+2h+1}
    const float* aPtr = &aLds[(rowBaseL + ln) * LDA + 2 * h];

    float expAcc[8];
#pragma unroll
    for (int r = 0; r < 8; ++r) expAcc[r] = 0.0f;

    for (int c = q; c < chunks; c += 4) {
        const int colBase = c << 4;
        // B-frag: B[k][n] = W[colBase+n][k]; lane ln = N, pair covers K = {4s+2h, 4s+2h+1}
        const float* bPtr = Wm + (size_t)(colBase + ln) * D + 2 * h;

        v8f acc = {};
#pragma unroll 16
        for (int s = 0; s < D / 4; ++s) {
            v2f a = *(const v2f*)(aPtr + 4 * s);     // ds_load_b64, conflict-free (pad=2)
            v2f b = *(const v2f*)(bPtr + 4 * s);     // global_load_b64, L2-resident W
            acc = __builtin_amdgcn_wmma_f32_16x16x4_f32(
                false, a, false, b, (short)0, acc, false, false);
        }

        // D-frag layout: VGPR r, half h -> row M = 8*h + r, col N = ln
#pragma unroll
        for (int r = 0; r < 8; ++r) {
            float logit = acc[r];
            expAcc[r] += __expf(S_SCALE * logit);
            int rl = rowBaseL + 8 * h + r;
            if (tgtIdx[rl] == colBase + ln)
                tgtVal[rl] = logit;                  // unique writer per row
        }
    }

    // ---- reduce exp sums across the 16 lanes of each N-group (wave32-safe) ----
#pragma unroll
    for (int r = 0; r < 8; ++r) {
        float v = expAcc[r];
        v += __shfl_xor(v, 1);
        v += __shfl_xor(v, 2);
        v += __shfl_xor(v, 4);
        v += __shfl_xor(v, 8);
        expAcc[r] = v;
    }
    if (ln == 0) {
#pragma unroll
        for (int r = 0; r < 8; ++r)
            rowSumP[q][rowBaseL + 8 * h + r] = expAcc[r];   // single writer per slot
    }
    __syncthreads();

    // ---- per-row arcface finalize (fixed-order sum => deterministic) ----
    if (t < ROWS) {
        float sum = rowSumP[0][t] + rowSumP[1][t] + rowSumP[2][t] + rowSumP[3][t];
        float tg  = tgtVal[t];
        float tc  = fminf(fmaxf(tg, -1.0f + 1e-7f), 1.0f - 1e-7f);
        float num = S_SCALE * cosf(acosf(tc) + MARGIN);
        float den = __expf(num) + (sum - __expf(S_SCALE * tg));
        rowOut[rowBaseG + t] = num - logf(den);
    }
}

// Deterministic single-block tree reduction: out = -mean(rowL)
__global__ __launch_bounds__(256)
void reduce_neg_mean(const float* __restrict__ rowL, float* __restrict__ out, int n)
{
    __shared__ float buf[256];
    int t = threadIdx.x;
    float s = 0.0f;
    for (int i = t; i < n; i += 256) s += rowL[i];
    buf[t] = s;
    __syncthreads();
    for (int w = 128; w > 0; w >>= 1) {
        if (t < w) buf[t] += buf[t + w];
        __syncthreads();
    }
    if (t == 0) out[0] = -buf[0] / (float)n;
}

extern "C" void kernel_launch(void* const* d_in, const int* in_sizes, int n_in,
                              void* d_out, int out_size, void* d_ws, size_t ws_size,
                              hipStream_t stream)
{
    const float* x      = (const float*)d_in[0];
    const float* Wm     = (const float*)d_in[1];
    const int*   target = (const int*)d_in[2];

    const int N = in_sizes[2];          // 8192
    const int C = in_sizes[1] / D;      // 10000

    float* rowL = (float*)d_ws;         // N floats of scratch

    arcface_fused<<<N / ROWS, 256, 0, stream>>>(x, Wm, target, rowL, C);
    reduce_neg_mean<<<1, 256, 0, stream>>>(rowL, (float*)d_out, N);
}